// MultiHeadSelfAttentionRoPE_36283883716794
// MI455X (gfx1250) — compile-verified
//
#include <hip/hip_runtime.h>
#include <math.h>

typedef __attribute__((ext_vector_type(2))) float v2f;
typedef __attribute__((ext_vector_type(8))) float v8f;

#define BATCH 2
#define SEQ 2048
#define DM 1024
#define NH 16
#define DK 64

// ---------- wave-level 32x64-tile GEMM:  out[m][n] = sum_k X[m][k] * W[n][k] ----------
// X: M x K row-major, W: N x K row-major.
// STORE_VT == 0: out[m][n] row-major M x N.
// STORE_VT == 1: out stored transposed per batch: out[(b*N + n)*SEQ + s], m = b*SEQ + s.
// One wave per 32x64 tile: 2 A-frags + 4 B-frags -> 8 independent WMMA chains.
template <int STORE_VT>
__global__ __launch_bounds__(256) void gemm_xwt_kernel(const float* __restrict__ X,
                                                       const float* __restrict__ W,
                                                       float* __restrict__ out,
                                                       int M, int N, int K) {
    const int lane = threadIdx.x & 31;
    const int warp = threadIdx.x >> 5;
    const int tile = blockIdx.x * 8 + warp;
    const int tilesN = N >> 6;               // 64-wide N tiles
    const int tm = (tile / tilesN) << 5;     // 32-tall M tiles
    const int tn = (tile % tilesN) << 6;
    const int ln = lane & 15;
    const int lh = lane >> 4;                // which K-half of the 16x4 fragment

    const float* xrow0 = X + (size_t)(tm + ln) * K + 2 * lh;
    const float* xrow1 = xrow0 + (size_t)16 * K;
    const float* wrow  = W + (size_t)(tn + ln) * K + 2 * lh;

    v8f c[2][4] = {};
#pragma unroll 4
    for (int k0 = 0; k0 < K; k0 += 4) {
        v2f a0 = *(const v2f*)(xrow0 + k0);
        v2f a1 = *(const v2f*)(xrow1 + k0);
        v2f b0 = *(const v2f*)(wrow + k0);
        v2f b1 = *(const v2f*)(wrow + (size_t)16 * K + k0);
        v2f b2 = *(const v2f*)(wrow + (size_t)32 * K + k0);
        v2f b3 = *(const v2f*)(wrow + (size_t)48 * K + k0);
        c[0][0] = __builtin_amdgcn_wmma_f32_16x16x4_f32(false, a0, false, b0, (short)0, c[0][0], false, false);
        c[0][1] = __builtin_amdgcn_wmma_f32_16x16x4_f32(false, a0, false, b1, (short)0, c[0][1], false, false);
        c[0][2] = __builtin_amdgcn_wmma_f32_16x16x4_f32(false, a0, false, b2, (short)0, c[0][2], false, false);
        c[0][3] = __builtin_amdgcn_wmma_f32_16x16x4_f32(false, a0, false, b3, (short)0, c[0][3], false, false);
        c[1][0] = __builtin_amdgcn_wmma_f32_16x16x4_f32(false, a1, false, b0, (short)0, c[1][0], false, false);
        c[1][1] = __builtin_amdgcn_wmma_f32_16x16x4_f32(false, a1, false, b1, (short)0, c[1][1], false, false);
        c[1][2] = __builtin_amdgcn_wmma_f32_16x16x4_f32(false, a1, false, b2, (short)0, c[1][2], false, false);
        c[1][3] = __builtin_amdgcn_wmma_f32_16x16x4_f32(false, a1, false, b3, (short)0, c[1][3], false, false);
    }

    if (STORE_VT == 0) {
#pragma unroll
        for (int mi = 0; mi < 2; ++mi)
#pragma unroll
            for (int ni = 0; ni < 4; ++ni) {
                float* orow = out + (size_t)(tm + mi * 16 + 8 * lh) * N + tn + ni * 16 + ln;
#pragma unroll
                for (int r = 0; r < 8; ++r)
                    orow[(size_t)r * N] = c[mi][ni][r];
            }
    } else {
        const int bb = tm >> 11;             // batch (SEQ == 2048, tiles never straddle)
#pragma unroll
        for (int mi = 0; mi < 2; ++mi)
#pragma unroll
            for (int ni = 0; ni < 4; ++ni) {
                int n = tn + ni * 16 + ln;
                int srow = (tm & (SEQ - 1)) + mi * 16 + 8 * lh;
                float* ocol = out + ((size_t)(bb * N + n) << 11) + srow;
#pragma unroll
                for (int r = 0; r < 8; ++r)
                    ocol[r] = c[mi][ni][r];  // contiguous along s per lane
            }
    }
}

// ---------- RoPE on Q and K, layout [b][s][h][dk] ----------
__global__ __launch_bounds__(256) void rope_kernel(float* __restrict__ q,
                                                   float* __restrict__ k, int total) {
    int tid = blockIdx.x * blockDim.x + threadIdx.x;
    if (tid >= total) return;
    int i = tid & 31;                  // pair/freq index
    int s = (tid >> 9) & (SEQ - 1);    // 16 heads * 32 pairs = 512 per (b,s)
    float freq = __expf(-(float)i * (1.0f / 32.0f) * 9.210340371976184f); // 10000^(-2i/64)
    float ang = (float)s * freq;
    float cs = __cosf(ang);
    float sn = __sinf(ang);
    size_t p = (size_t)tid * 2;
    float q1 = q[p], q2 = q[p + 1];
    q[p]     = cs * q1 - sn * q2;
    q[p + 1] = sn * q1 + cs * q2;
    float k1 = k[p], k2 = k[p + 1];
    k[p]     = cs * k1 - sn * k2;
    k[p + 1] = sn * k1 + cs * k2;
}

// ---------- row reductions across the 16 lanes of a half-wave ----------
__device__ __forceinline__ float rowmax16(float v) {
    v = fmaxf(v, __shfl_xor(v, 1, 32));
    v = fmaxf(v, __shfl_xor(v, 2, 32));
    v = fmaxf(v, __shfl_xor(v, 4, 32));
    v = fmaxf(v, __shfl_xor(v, 8, 32));
    return v;
}
__device__ __forceinline__ float rowsum16(float v) {
    v += __shfl_xor(v, 1, 32);
    v += __shfl_xor(v, 2, 32);
    v += __shfl_xor(v, 4, 32);
    v += __shfl_xor(v, 8, 32);
    return v;
}

// ---------- flash attention, causal, one wave per (b, h, 16-query tile) ----------
// q,k layout: [b][s][h*64+dk], row stride DM.  vt layout: [b][n][s], n = h*64+dk.
__global__ __launch_bounds__(256) void attn_kernel(const float* __restrict__ q,
                                                   const float* __restrict__ k,
                                                   const float* __restrict__ vt,
                                                   float* __restrict__ o) {
    __shared__ float lds[8][16 * 18];          // per-wave padded P tile (stride 18)
    const int lane = threadIdx.x & 31;
    const int warp = threadIdx.x >> 5;
    const int wave = blockIdx.x * 8 + warp;
    const int qt = wave & 127;                 // query tile 0..127
    const int bh = wave >> 7;
    const int h = bh & 15;
    const int b = bh >> 4;
    const int ln = lane & 15;
    const int lh = lane >> 4;

    const float* qbase = q + ((size_t)(b * SEQ) + qt * 16 + ln) * DM + h * DK;
    v2f qa[16];
#pragma unroll
    for (int j = 0; j < 16; ++j)
        qa[j] = *(const v2f*)(qbase + 4 * j + 2 * lh);

    // Per-lane Vt row pointers: lane covers feature n = h*64 + nt*16 + ln.
    const float* vrow[4];
#pragma unroll
    for (int nt = 0; nt < 4; ++nt)
        vrow[nt] = vt + ((size_t)(b * DM + h * DK + nt * 16 + ln) << 11);

    v8f acc[4] = {v8f{}, v8f{}, v8f{}, v8f{}};
    float mrow[8], lrow[8];
#pragma unroll
    for (int r = 0; r < 8; ++r) { mrow[r] = -1e30f; lrow[r] = 0.0f; }

    float* pl = &lds[warp][0];

    for (int kt = 0; kt <= qt; ++kt) {
        // ---- scores tile S = (Q K^T) * 1/sqrt(dk), 16 chained f32 WMMAs ----
        const float* kbase = k + ((size_t)(b * SEQ) + kt * 16 + ln) * DM + h * DK;
        v8f s = {};
#pragma unroll
        for (int j = 0; j < 16; ++j) {
            v2f bf = *(const v2f*)(kbase + 4 * j + 2 * lh);
            s = __builtin_amdgcn_wmma_f32_16x16x4_f32(false, qa[j], false, bf,
                                                      (short)0, s, false, false);
        }
        // ---- mask + online softmax ----
        float nm[8], rs[8];
#pragma unroll
        for (int r = 0; r < 8; ++r) {
            float sv = s[r] * 0.125f;                         // 1/sqrt(64)
            if (kt == qt && ln > (r + 8 * lh)) sv = -1e30f;   // causal within diag tile
            s[r] = sv;
            nm[r] = fmaxf(mrow[r], rowmax16(sv));
        }
#pragma unroll
        for (int r = 0; r < 8; ++r) {
            float p = __expf(s[r] - nm[r]);
            s[r] = p;
            rs[r] = rowsum16(p);
        }
#pragma unroll
        for (int r = 0; r < 8; ++r) {
            float sc = __expf(mrow[r] - nm[r]);
            lrow[r] = lrow[r] * sc + rs[r];
            mrow[r] = nm[r];
#pragma unroll
            for (int nt = 0; nt < 4; ++nt) acc[nt][r] *= sc;
        }
        // ---- transpose P (C-layout) -> A-layout via per-wave LDS tile ----
#pragma unroll
        for (int r = 0; r < 8; ++r)
            pl[(r + 8 * lh) * 18 + ln] = s[r];
        // ---- acc += P @ V  (16 WMMAs: 4 k-steps x 4 output col-tiles) ----
#pragma unroll
        for (int j = 0; j < 4; ++j) {
            int kk = 4 * j + 2 * lh;
            v2f a = *(const v2f*)(pl + ln * 18 + kk);         // 8B-aligned ds_load_b64
#pragma unroll
            for (int nt = 0; nt < 4; ++nt) {
                v2f bf = *(const v2f*)(vrow[nt] + kt * 16 + kk);  // consecutive keys along s
                acc[nt] = __builtin_amdgcn_wmma_f32_16x16x4_f32(false, a, false, bf,
                                                                (short)0, acc[nt],
                                                                false, false);
            }
        }
    }

    // ---- epilogue: divide by softmax denom, write [b][s][h*64+dk] ----
    float* obase = o + ((size_t)(b * SEQ) + qt * 16) * DM + h * DK;
#pragma unroll
    for (int r = 0; r < 8; ++r) {
        float inv = 1.0f / lrow[r];
        int m = r + 8 * lh;
#pragma unroll
        for (int nt = 0; nt < 4; ++nt)
            obase[(size_t)m * DM + nt * 16 + ln] = acc[nt][r] * inv;
    }
}

extern "C" void kernel_launch(void* const* d_in, const int* in_sizes, int n_in,
                              void* d_out, int out_size, void* d_ws, size_t ws_size,
                              hipStream_t stream) {
    const float* x  = (const float*)d_in[0];
    const float* Wq = (const float*)d_in[1];
    const float* Wk = (const float*)d_in[2];
    const float* Wv = (const float*)d_in[3];
    const float* Wo = (const float*)d_in[4];
    float* out = (float*)d_out;

    const size_t bufElems = (size_t)BATCH * SEQ * DM;   // 4M floats = 16 MB
    float* qb  = (float*)d_ws;
    float* kb  = qb + bufElems;
    float* vtb = kb + bufElems;                         // V stored transposed [b][n][s]
    float* ab  = vtb + bufElems;

    const int M = BATCH * SEQ, N = DM, K = DM;
    const int tiles = (M / 32) * (N / 64);              // 2048 tiles of 32x64
    const int gblocks = tiles / 8;                      // 256 blocks of 8 waves

    gemm_xwt_kernel<0><<<gblocks, 256, 0, stream>>>(x, Wq, qb, M, N, K);
    gemm_xwt_kernel<0><<<gblocks, 256, 0, stream>>>(x, Wk, kb, M, N, K);
    gemm_xwt_kernel<1><<<gblocks, 256, 0, stream>>>(x, Wv, vtb, M, N, K);

    const int pairs = BATCH * SEQ * NH * (DK / 2);      // 2,097,152
    rope_kernel<<<pairs / 256, 256, 0, stream>>>(qb, kb, pairs);

    const int waves = BATCH * NH * (SEQ / 16);          // 4096 waves
    attn_kernel<<<waves / 8, 256, 0, stream>>>(qb, kb, vtb, ab);

    gemm_xwt_kernel<0><<<gblocks, 256, 0, stream>>>(ab, Wo, out, M, N, K);
}